// GCAGNN_14843406975360
// MI455X (gfx1250) — compile-verified
//
#include <hip/hip_runtime.h>
#include <cstdint>

typedef float v2f __attribute__((ext_vector_type(2)));
typedef float v8f __attribute__((ext_vector_type(8)));

#define A_STRIDE 516   // 16 rows x (max K=512) padded: 516 % 64 == 4 -> conflict-free
#define H_STRIDE 260   // 16 rows x 256 padded

// ---------------- Cl(0,1,1,1) blade product (bitmask form) ----------------
// blade index (grade-lex order) -> bitmask over {e0,e1,e2,e3}
__device__ __constant__ int IDX2MASK[16] = {
    0x0, 0x1, 0x2, 0x4, 0x8, 0x3, 0x5, 0x9, 0x6, 0xA, 0xC, 0x7, 0xB, 0xD, 0xE, 0xF};
// action blades: scalar, 6 bivectors, pseudoscalar
__device__ __constant__ int AMASK[8] = {0x0, 0x3, 0x5, 0x9, 0x6, 0xA, 0xC, 0xF};
__device__ __constant__ float AREV[8] = {1.f, -1.f, -1.f, -1.f, -1.f, -1.f, -1.f, 1.f};

__device__ __forceinline__ float blade_mul(int ma, int mb, int* mk) {
    int res = ma, swaps = 0;
    float coeff = 1.0f;
    for (int g = 0; g < 4; ++g) {
        if (mb & (1 << g)) {
            swaps += __popc(res >> (g + 1));          // reordering sign
            if (res & (1 << g)) {
                if (g == 0) coeff = 0.0f;             // e0^2 = 0 (degenerate)
                res &= ~(1 << g);
            } else {
                res |= (1 << g);
            }
        }
    }
    *mk = res;
    return (swaps & 1) ? -coeff : coeff;
}

// ---------------- W_eff builder: W[(i,q),(o,k)] = w[o,i] * (a e_q rev(a))_k ----------------
// Stored K-pair interleaved: wpk[((kk>>1)*Nout + col)*2 + (kk&1)], kk=i*16+q, col=o*16+k,
// so a WMMA B-fragment lane load is one aligned float2.
__global__ void build_weff(const float* __restrict__ w, const float* __restrict__ a8,
                           float* __restrict__ wpk, int O, int I) {
    int oi = blockIdx.x;                 // o*I + i
    int i  = oi % I;
    int q  = threadIdx.x >> 4;
    int k  = threadIdx.x & 15;
    float av[8];
#pragma unroll
    for (int p = 0; p < 8; ++p) av[p] = a8[(size_t)oi * 8 + p];
    int qm = IDX2MASK[q], km = IDX2MASK[k];
    float acc = 0.f;
#pragma unroll
    for (int p = 0; p < 8; ++p) {
        int rm;
        float s1 = blade_mul(AMASK[p], qm, &rm);     // e_p * e_q -> s1 * e_rm
        if (s1 != 0.f) {
#pragma unroll
            for (int s = 0; s < 8; ++s) {
                int km2;
                float s2 = blade_mul(rm, AMASK[s], &km2);  // e_rm * e_s -> s2 * e_km2
                if (km2 == km) acc += av[p] * s1 * (av[s] * AREV[s]) * s2;
            }
        }
    }
    float weff = w[oi] * acc;
    int kk   = i * 16 + q;
    int col  = (oi / I) * 16 + k;
    int Nout = O * 16;
    wpk[((size_t)(kk >> 1) * Nout + col) * 2 + (kk & 1)] = weff;
}

// ---------------- WMMA helpers (f32 16x16x4) ----------------
// A frag: lane l holds (M=l&15, K=kb+2*(l>>4)+{0,1}); B frag: (K=kb+2*(l>>4)+{0,1}, N=l&15).
// Software-pipelined: fragments for step kb+4 are loaded before the WMMAs of step kb issue,
// so the matrix ops never wait on the current iteration's memory.
template <int NT>
__device__ __forceinline__ void gemm_phase(const float* __restrict__ src, int srcStride, int K,
                                           const float2* __restrict__ Bpk, int Nout, int wtb,
                                           v8f (&acc)[NT], int lane) {
    int m = lane & 15, half = lane >> 4;
#pragma unroll
    for (int t = 0; t < NT; ++t)
#pragma unroll
        for (int j = 0; j < 8; ++j) acc[t][j] = 0.f;
    const float* arow = src + m * srcStride + 2 * half;
    const float2* bcol = Bpk + (size_t)half * Nout + (wtb + m);

    v2f a = *(const v2f*)(arow);
    float2 b[NT];
#pragma unroll
    for (int t = 0; t < NT; ++t) b[t] = bcol[t * 16];

    for (int kb = 0; kb + 4 < K; kb += 4) {
        v2f a_n = *(const v2f*)(arow + kb + 4);
        const float2* bn = bcol + (size_t)((kb + 4) >> 1) * Nout;
        float2 b_n[NT];
#pragma unroll
        for (int t = 0; t < NT; ++t) b_n[t] = bn[t * 16];
#pragma unroll
        for (int t = 0; t < NT; ++t) {
            v2f bb = {b[t].x, b[t].y};
            acc[t] = __builtin_amdgcn_wmma_f32_16x16x4_f32(
                false, a, false, bb, (short)0, acc[t], false, false);
        }
        a = a_n;
#pragma unroll
        for (int t = 0; t < NT; ++t) b[t] = b_n[t];
    }
#pragma unroll
    for (int t = 0; t < NT; ++t) {
        v2f bb = {b[t].x, b[t].y};
        acc[t] = __builtin_amdgcn_wmma_f32_16x16x4_f32(
            false, a, false, bb, (short)0, acc[t], false, false);
    }
}

// 16-lane XOR butterfly via ds_swizzle (group-of-32: offset = xor<<10 | and_mask 0x1F)
__device__ __forceinline__ float swz_add1(float x) {
    return x + __int_as_float(__builtin_amdgcn_ds_swizzle(__float_as_int(x), 0x041F));
}
__device__ __forceinline__ float swz_add2(float x) {
    return x + __int_as_float(__builtin_amdgcn_ds_swizzle(__float_as_int(x), 0x081F));
}
__device__ __forceinline__ float swz_add4(float x) {
    return x + __int_as_float(__builtin_amdgcn_ds_swizzle(__float_as_int(x), 0x101F));
}
__device__ __forceinline__ float swz_add8(float x) {
    return x + __int_as_float(__builtin_amdgcn_ds_swizzle(__float_as_int(x), 0x201F));
}

// gate over one 16-col tile == one multivector channel c: g=sigmoid(sum_m D[r,m]*kw[c,m]+kb[c])
__device__ __forceinline__ void gate_tile(v8f& acc, const float* __restrict__ kw,
                                          const float* __restrict__ kb, int c, int lane) {
    int m = lane & 15;
    float kwv = kw[c * 16 + m];
    float kbv = kb[c];
#pragma unroll
    for (int v = 0; v < 8; ++v) {
        float t = acc[v] * kwv;
        t = swz_add1(t);
        t = swz_add2(t);
        t = swz_add4(t);
        t = swz_add8(t);                  // 16-lane row sum within each half
        float g = 1.f / (1.f + __expf(-(t + kbv)));
        acc[v] *= g;
    }
}

__device__ __forceinline__ void store_tile(float* __restrict__ dst, const v8f& acc, int tileBase,
                                           int lane) {
    int m = lane & 15, half = lane >> 4;
#pragma unroll
    for (int v = 0; v < 8; ++v) dst[(v + 8 * half) * H_STRIDE + tileBase + m] = acc[v];
}

// ---------------- fused message MLP + scatter-add ----------------
__global__ __launch_bounds__(128) void msg_kernel(
    const float* __restrict__ xin, const int* __restrict__ ei, int E, int C16,
    const float2* __restrict__ W1, const float2* __restrict__ W2, const float2* __restrict__ W3,
    const float* __restrict__ kw0, const float* __restrict__ kb0,
    const float* __restrict__ kw1, const float* __restrict__ kb1,
    float* __restrict__ agg) {
    __shared__ float Abuf[16 * A_STRIDE];
    __shared__ float Hbuf[16 * H_STRIDE];
    __shared__ int sDst[16], sSrc[16];
    int tid = threadIdx.x, lane = tid & 31, wave = tid >> 5;
    int e0 = blockIdx.x * 16;
    if (tid < 16) {
        sSrc[tid] = ei[e0 + tid];       // edge_index[0] = src
        sDst[tid] = ei[E + e0 + tid];   // edge_index[1] = dst
    }
    __syncthreads();
    int segw = C16 >> 1;            // float2s per node row
    int roww = segw * 2;            // float2s per feat row (concat dst,src)
    for (int idx = tid; idx < 16 * roww; idx += 128) {
        int r = idx / roww, off = idx - r * roww;
        int node = (off < segw) ? sDst[r] : sSrc[r];
        int seg  = (off < segw) ? off : off - segw;
        float2 v = ((const float2*)(xin + (size_t)node * C16))[seg];
        ((float2*)(Abuf + r * A_STRIDE))[off] = v;
    }
    __syncthreads();

    v8f acc[4];
    int wtb = wave * 64;
    // GEMM1 + act0
    gemm_phase<4>(Abuf, A_STRIDE, 2 * C16, W1, 256, wtb, acc, lane);
#pragma unroll
    for (int t = 0; t < 4; ++t) gate_tile(acc[t], kw0, kb0, wave * 4 + t, lane);
#pragma unroll
    for (int t = 0; t < 4; ++t) store_tile(Hbuf, acc[t], wtb + t * 16, lane);
    __syncthreads();
    // GEMM2 + act1
    gemm_phase<4>(Hbuf, H_STRIDE, 256, W2, 256, wtb, acc, lane);
#pragma unroll
    for (int t = 0; t < 4; ++t) gate_tile(acc[t], kw1, kb1, wave * 4 + t, lane);
    __syncthreads();
#pragma unroll
    for (int t = 0; t < 4; ++t) store_tile(Hbuf, acc[t], wtb + t * 16, lane);
    __syncthreads();
    // GEMM3 -> scatter-add into agg[dst]
    gemm_phase<4>(Hbuf, H_STRIDE, 256, W3, 256, wtb, acc, lane);
    int m = lane & 15, half = lane >> 4;
#pragma unroll
    for (int t = 0; t < 4; ++t) {
        int col = wtb + t * 16 + m;
#pragma unroll
        for (int v = 0; v < 8; ++v) {
            int r = v + 8 * half;
            atomicAdd(&agg[(size_t)sDst[r] * 256 + col], acc[t][v]);
        }
    }
}

// ---------------- fused update MLP ----------------
__global__ __launch_bounds__(128) void upd_kernel(
    const float* __restrict__ xin, const float* __restrict__ agg, int C16,
    const float2* __restrict__ W1, const float2* __restrict__ W2, const float2* __restrict__ W3,
    const float* __restrict__ kw0, const float* __restrict__ kb0,
    const float* __restrict__ kw1, const float* __restrict__ kb1,
    float* __restrict__ xout, int Nout) {
    __shared__ float Abuf[16 * A_STRIDE];
    __shared__ float Hbuf[16 * H_STRIDE];
    int tid = threadIdx.x, lane = tid & 31, wave = tid >> 5;
    int n0 = blockIdx.x * 16;
    int Kin = C16 + 256;
    int xw = C16 >> 1, roww = Kin >> 1;
    for (int idx = tid; idx < 16 * roww; idx += 128) {
        int r = idx / roww, off = idx - r * roww;
        int n = n0 + r;
        float2 v = (off < xw) ? ((const float2*)(xin + (size_t)n * C16))[off]
                              : ((const float2*)(agg + (size_t)n * 256))[off - xw];
        ((float2*)(Abuf + r * A_STRIDE))[off] = v;
    }
    __syncthreads();

    v8f acc[4];
    int wtb = wave * 64;
    gemm_phase<4>(Abuf, A_STRIDE, Kin, W1, 256, wtb, acc, lane);
#pragma unroll
    for (int t = 0; t < 4; ++t) gate_tile(acc[t], kw0, kb0, wave * 4 + t, lane);
#pragma unroll
    for (int t = 0; t < 4; ++t) store_tile(Hbuf, acc[t], wtb + t * 16, lane);
    __syncthreads();
    gemm_phase<4>(Hbuf, H_STRIDE, 256, W2, 256, wtb, acc, lane);
#pragma unroll
    for (int t = 0; t < 4; ++t) gate_tile(acc[t], kw1, kb1, wave * 4 + t, lane);
    __syncthreads();
#pragma unroll
    for (int t = 0; t < 4; ++t) store_tile(Hbuf, acc[t], wtb + t * 16, lane);
    __syncthreads();

    int m = lane & 15, half = lane >> 4;
    if (Nout == 256) {
        gemm_phase<4>(Hbuf, H_STRIDE, 256, W3, 256, wave * 64, acc, lane);
#pragma unroll
        for (int t = 0; t < 4; ++t) {
            int col = wave * 64 + t * 16 + m;
#pragma unroll
            for (int v = 0; v < 8; ++v)
                xout[(size_t)(n0 + v + 8 * half) * 256 + col] = acc[t][v];
        }
    } else {  // Nout == 128 (final layer, 8 out channels)
        v8f acc2[2];
        gemm_phase<2>(Hbuf, H_STRIDE, 256, W3, 128, wave * 32, acc2, lane);
#pragma unroll
        for (int t = 0; t < 2; ++t) {
            int col = wave * 32 + t * 16 + m;
#pragma unroll
            for (int v = 0; v < 8; ++v)
                xout[(size_t)(n0 + v + 8 * half) * 128 + col] = acc2[t][v];
        }
    }
}

__global__ void zero_kernel(float* __restrict__ p, int n) {
    int i = blockIdx.x * 256 + threadIdx.x;
    if (i < n) p[i] = 0.f;
}

// ---------------- launcher ----------------
extern "C" void kernel_launch(void* const* d_in, const int* in_sizes, int n_in,
                              void* d_out, int out_size, void* d_ws, size_t ws_size,
                              hipStream_t stream) {
    (void)n_in; (void)out_size; (void)ws_size;
    const float* x  = (const float*)d_in[0];
    const int*   ei = (const int*)d_in[1];
    const int N = in_sizes[0] / 128;   // (N, 8, 16) fp32
    const int E = in_sizes[1] / 2;     // (2, E) int32

    // Param flattening (insertion order, depth-first):
    // per layer: message{lin0.w,lin0.a,lin1.w,lin1.a,lin2.w,lin2.a,act0.kw,act0.kb,act1.kw,act1.kb}
    //            then update{same 10}
    const int Cin[3]  = {8, 16, 16};
    const int Cout[3] = {16, 16, 8};

    float* wsf = (float*)d_ws;
    size_t off = 0;
    float* Wm[3][3];
    float* Wu[3][3];
    for (int l = 0; l < 3; ++l) {
        int base = 2 + l * 20;
        int mi[3] = {2 * Cin[l], 16, 16}, mo[3] = {16, 16, 16};
        int ui[3] = {Cin[l] + 16, 16, 16}, uo[3] = {16, 16, Cout[l]};
        for (int j = 0; j < 3; ++j) {
            Wm[l][j] = wsf + off;
            off += (size_t)mi[j] * 16 * mo[j] * 16;
            build_weff<<<mo[j] * mi[j], 256, 0, stream>>>(
                (const float*)d_in[base + 2 * j], (const float*)d_in[base + 2 * j + 1],
                Wm[l][j], mo[j], mi[j]);
        }
        for (int j = 0; j < 3; ++j) {
            Wu[l][j] = wsf + off;
            off += (size_t)ui[j] * 16 * uo[j] * 16;
            build_weff<<<uo[j] * ui[j], 256, 0, stream>>>(
                (const float*)d_in[base + 10 + 2 * j], (const float*)d_in[base + 10 + 2 * j + 1],
                Wu[l][j], uo[j], ui[j]);
        }
    }

    float* agg = wsf + off; off += (size_t)N * 256;
    float* xA  = wsf + off; off += (size_t)N * 256;
    float* xB  = wsf + off; off += (size_t)N * 256;

    const float* xcur = x;
    int C = 8;
    for (int l = 0; l < 3; ++l) {
        int base = 2 + l * 20;
        int aggN = N * 256;
        zero_kernel<<<(aggN + 255) / 256, 256, 0, stream>>>(agg, aggN);
        msg_kernel<<<E / 16, 128, 0, stream>>>(
            xcur, ei, E, C * 16,
            (const float2*)Wm[l][0], (const float2*)Wm[l][1], (const float2*)Wm[l][2],
            (const float*)d_in[base + 6], (const float*)d_in[base + 7],
            (const float*)d_in[base + 8], (const float*)d_in[base + 9], agg);
        float* xnext = (l == 2) ? (float*)d_out : (l == 0 ? xA : xB);
        upd_kernel<<<N / 16, 128, 0, stream>>>(
            xcur, agg, C * 16,
            (const float2*)Wu[l][0], (const float2*)Wu[l][1], (const float2*)Wu[l][2],
            (const float*)d_in[base + 16], (const float*)d_in[base + 17],
            (const float*)d_in[base + 18], (const float*)d_in[base + 19],
            xnext, Cout[l] * 16);
        xcur = xnext;
        C = Cout[l];
    }
}